// fast_weights_model_5720896438740
// MI455X (gfx1250) — compile-verified
//
#include <hip/hip_runtime.h>
#include <hip/hip_bf16.h>

typedef float v2f __attribute__((ext_vector_type(2)));
typedef float v8f __attribute__((ext_vector_type(8)));

#define B_N   2048
#define T_N   50
#define DIN_N 128
#define D1_N  256
#define H_N   64
#define C_N   64
#define LAM_C 0.95f
#define ETA_C 0.5f
#define MB    32                 // batch rows per block
#define NBLK  (B_N / MB)         // 64 persistent blocks
#define NTHR  256                // 8 waves (wave32)
#define NWAVE 8
#define S1_LD (D1_N + 4)         // 260: 16-row strided LDS reads hit distinct banks
#define H_LD  (H_N + 4)          // 68

// ---- exact-f32 WMMA: D = A(16x4) * B(4x16) + C, one K=4 slab per issue ----
__device__ __forceinline__ v8f wmma4(v8f acc, v2f a, v2f b) {
  return __builtin_amdgcn_wmma_f32_16x16x4_f32(false, a, false, b, (short)0, acc,
                                               false, false);
}

__device__ __forceinline__ v8f fill8(float x) {
  v8f r;
#pragma unroll
  for (int i = 0; i < 8; ++i) r[i] = x;
  return r;
}

// 16x16 output tile GEMM step. A: row-major tile base (lda), B: row-major col base (ldb).
// A-frag layout (32-bit A 16x4): lane<16 -> K={k0,k0+1}, lane>=16 -> K={k0+2,k0+3}, M=lane&15
// B-frag layout (rows striped like C): vgpr v, half h -> K = k0 + v + 2*h, N = lane&15
__device__ __forceinline__ v8f gemm_tile(const float* A, int lda,
                                         const float* Bm, int ldb,
                                         int K, v8f acc, int lane) {
  const int half = lane >> 4;
  const int l = lane & 15;
  const float* ap = A + (size_t)l * lda + 2 * half;
  const float* bp = Bm + l + (size_t)(2 * half) * ldb;
  for (int k0 = 0; k0 < K; k0 += 4) {
    v2f av, bv;
    av.x = ap[k0];
    av.y = ap[k0 + 1];
    bv.x = bp[(size_t)k0 * ldb];
    bv.y = bp[(size_t)(k0 + 1) * ldb];
    acc = wmma4(acc, av, bv);
  }
  return acc;
}

// C/D layout: vgpr i, half h -> M = i + 8*h, N = lane&15
__device__ __forceinline__ void store_tile(float* Cp, int ldc, v8f acc, int lane,
                                           bool relu) {
  const int half = lane >> 4;
  const int l = lane & 15;
  float* cp = Cp + (size_t)(8 * half) * ldc + l;
#pragma unroll
  for (int i = 0; i < 8; ++i) {
    float v = acc[i];
    if (relu) v = fmaxf(v, 0.0f);
    cp[(size_t)i * ldc] = v;
  }
}

// two-variable sense-reversal grid barrier (64 co-resident blocks)
__device__ __forceinline__ void grid_sync(unsigned* cnt, unsigned* gen, int nblk) {
  __threadfence();
  __syncthreads();
  if (threadIdx.x == 0) {
    unsigned g = atomicAdd(gen, 0u);
    if (atomicAdd(cnt, 1u) == (unsigned)(nblk - 1)) {
      atomicExch(cnt, 0u);
      atomicAdd(gen, 1u);
    } else {
      while (atomicAdd(gen, 0u) == g) __builtin_amdgcn_s_sleep(2);
    }
  }
  __syncthreads();
  __threadfence();
}

__global__ void fw_init(unsigned* bar, float* out) {
  if (threadIdx.x == 0) {
    bar[0] = 0u;
    bar[1] = 0u;
    out[0] = 0.0f;
    out[1] = 0.0f;
  }
}

__global__ __launch_bounds__(NTHR) void fw_main(
    const float* __restrict__ bx, const float* __restrict__ by,
    const float* __restrict__ w1, const float* __restrict__ b1,
    const float* __restrict__ w2, const float* __restrict__ b2,
    const float* __restrict__ wr, const float* __restrict__ cm,
    const float* __restrict__ gv, const float* __restrict__ bvv,
    const float* __restrict__ w3, const float* __restrict__ b3,
    const float* __restrict__ w4, const float* __restrict__ b4,
    float* __restrict__ a_state,   // [B,64,64] f32, L2-resident working set
    float* __restrict__ partial,   // [2][NBLK][128] LN partial sums (double buffered)
    unsigned* bar,                 // [2] barrier counter + generation
    float* __restrict__ out)       // out[0]=loss, out[1]=acc
{
  __shared__ float s1[MB][S1_LD];     // 32x260 ; also reused: hs overlays it, h3 reuses it
  __shared__ float zs[MB][H_LD];      // z (and final logits)
  __shared__ float hn[MB][H_LD];      // h after recurrent relu
  __shared__ float hloc[MB][H_LD];    // carry h across all 50 steps (never leaves LDS)
  __shared__ float mu[H_N];
  __shared__ float rsig[H_N];
  float (*hs)[H_LD] = (float(*)[H_LD])(&s1[0][0]);  // alias (disjoint in time)

  const int tid = threadIdx.x;
  const int lane = tid & 31;
  const int wv = tid >> 5;
  const int b0 = blockIdx.x * MB;

  // zero this block's private fast-weight slab (re-zeroed every call: deterministic)
  {
    float4* ap = (float4*)(a_state + (size_t)b0 * (H_N * H_N));
    for (int i = tid; i < MB * H_N * H_N / 4; i += NTHR)
      ap[i] = make_float4(0.f, 0.f, 0.f, 0.f);
  }
  for (int i = tid; i < MB * H_LD; i += NTHR) (&hloc[0][0])[i] = 0.f;
  __syncthreads();

  for (int t = 0; t < T_N; ++t) {
    // ---- s1 = relu(x_t @ w1 + b1): [32,128]x[128,256] = 32 tiles, 4 per wave
    for (int tile = wv; tile < 2 * (D1_N / 16); tile += NWAVE) {
      const int tm = tile >> 4, tn = tile & 15;
      v8f acc = fill8(b1[tn * 16 + (lane & 15)]);
      acc = gemm_tile(bx + ((size_t)(b0 + tm * 16) * T_N + t) * DIN_N, T_N * DIN_N,
                      w1 + tn * 16, D1_N, DIN_N, acc, lane);
      store_tile(&s1[tm * 16][tn * 16], S1_LD, acc, lane, true);
    }
    __syncthreads();
    // ---- z = relu(s1 @ w2 + b2): [32,256]x[256,64] = 8 tiles, 1 per wave
    {
      const int tm = wv >> 2, tn = wv & 3;
      v8f acc = fill8(b2[tn * 16 + (lane & 15)]);
      acc = gemm_tile(&s1[tm * 16][0], S1_LD, w2 + tn * 16, H_N, D1_N, acc, lane);
      store_tile(&zs[tm * 16][tn * 16], H_LD, acc, lane, true);
    }
    __syncthreads();
    // ---- hn = relu(h_old @ w + z @ c)
    {
      const int tm = wv >> 2, tn = wv & 3;
      v8f acc = fill8(0.f);
      acc = gemm_tile(&hloc[tm * 16][0], H_LD, wr + tn * 16, H_N, H_N, acc, lane);
      acc = gemm_tile(&zs[tm * 16][0], H_LD, cm + tn * 16, H_N, H_N, acc, lane);
      store_tile(&hn[tm * 16][tn * 16], H_LD, acc, lane, true);
    }
    __syncthreads();
    // ---- hs_pre = hn @ w + z @ c   (stored into alias of s1, which is now dead)
    {
      const int tm = wv >> 2, tn = wv & 3;
      v8f acc = fill8(0.f);
      acc = gemm_tile(&hn[tm * 16][0], H_LD, wr + tn * 16, H_N, H_N, acc, lane);
      acc = gemm_tile(&zs[tm * 16][0], H_LD, cm + tn * 16, H_N, H_N, acc, lane);
      store_tile(&hs[tm * 16][tn * 16], H_LD, acc, lane, false);
    }
    __syncthreads();
    // ---- fused fast-weight pass: a = LAM*a + ETA*h h^T, hs += h @ a_new
    // one streaming read+write of the 16KB/row slab; 8 threads per row, 8 cols each
    {
      const int row = tid >> 3;
      const int j0 = (tid & 7) * 8;
      float hj[8], m[8];
#pragma unroll
      for (int k = 0; k < 8; ++k) { hj[k] = hn[row][j0 + k]; m[k] = 0.f; }
      float4* ap = (float4*)(a_state + (size_t)(b0 + row) * (H_N * H_N) + j0);
      for (int i = 0; i < H_N; ++i) {
        const float hi = hn[row][i];
        const float eh = ETA_C * hi;
        float4 a0 = ap[(size_t)i * (H_N / 4)];
        float4 a1 = ap[(size_t)i * (H_N / 4) + 1];
        a0.x = fmaf(eh, hj[0], LAM_C * a0.x); m[0] = fmaf(hi, a0.x, m[0]);
        a0.y = fmaf(eh, hj[1], LAM_C * a0.y); m[1] = fmaf(hi, a0.y, m[1]);
        a0.z = fmaf(eh, hj[2], LAM_C * a0.z); m[2] = fmaf(hi, a0.z, m[2]);
        a0.w = fmaf(eh, hj[3], LAM_C * a0.w); m[3] = fmaf(hi, a0.w, m[3]);
        a1.x = fmaf(eh, hj[4], LAM_C * a1.x); m[4] = fmaf(hi, a1.x, m[4]);
        a1.y = fmaf(eh, hj[5], LAM_C * a1.y); m[5] = fmaf(hi, a1.y, m[5]);
        a1.z = fmaf(eh, hj[6], LAM_C * a1.z); m[6] = fmaf(hi, a1.z, m[6]);
        a1.w = fmaf(eh, hj[7], LAM_C * a1.w); m[7] = fmaf(hi, a1.w, m[7]);
        ap[(size_t)i * (H_N / 4)] = a0;
        ap[(size_t)i * (H_N / 4) + 1] = a1;
      }
#pragma unroll
      for (int k = 0; k < 8; ++k) hs[row][j0 + k] += m[k];
    }
    __syncthreads();
    // ---- per-j partial sums for batch-dim layernorm (no atomics)
    if (tid < H_N) {
      float s = 0.f, sq = 0.f;
      for (int r = 0; r < MB; ++r) {
        const float v = hs[r][tid];
        s += v;
        sq = fmaf(v, v, sq);
      }
      float* pp = partial + ((size_t)(t & 1) * NBLK + blockIdx.x) * 128;
      pp[tid] = s;
      pp[64 + tid] = sq;
    }
    grid_sync(bar, bar + 1, NBLK);   // one grid-wide sync per timestep
    // ---- LN stats over the whole batch
    if (tid < H_N) {
      float s = 0.f, sq = 0.f;
      const float* pb = partial + (size_t)(t & 1) * NBLK * 128;
      for (int blk = 0; blk < NBLK; ++blk) {
        s += pb[blk * 128 + tid];
        sq += pb[blk * 128 + 64 + tid];
      }
      const float m = s * (1.0f / B_N);
      mu[tid] = m;
      rsig[tid] = rsqrtf(fmaf(-m, m, sq * (1.0f / B_N)));
    }
    __syncthreads();
    // ---- h = relu(g*(hs-mu)/sig + b)  (stays in LDS for next step)
    {
      const int row = tid >> 3;
      const int j0 = (tid & 7) * 8;
#pragma unroll
      for (int k = 0; k < 8; ++k) {
        const int j = j0 + k;
        hloc[row][j] =
            fmaxf(fmaf(gv[j], (hs[row][j] - mu[j]) * rsig[j], bvv[j]), 0.f);
      }
    }
    __syncthreads();   // protect hs/s1 alias before next step overwrites it
  }

  // ---- head: h3 = relu(h @ w3 + b3) -> reuse s1
  for (int tile = wv; tile < 2 * (D1_N / 16); tile += NWAVE) {
    const int tm = tile >> 4, tn = tile & 15;
    v8f acc = fill8(b3[tn * 16 + (lane & 15)]);
    acc = gemm_tile(&hloc[tm * 16][0], H_LD, w3 + tn * 16, D1_N, H_N, acc, lane);
    store_tile(&s1[tm * 16][tn * 16], S1_LD, acc, lane, true);
  }
  __syncthreads();
  // ---- logits = h3 @ w4 + b4 -> reuse zs
  {
    const int tm = wv >> 2, tn = wv & 3;
    v8f acc = fill8(b4[tn * 16 + (lane & 15)]);
    acc = gemm_tile(&s1[tm * 16][0], S1_LD, w4 + tn * 16, C_N, D1_N, acc, lane);
    store_tile(&zs[tm * 16][tn * 16], H_LD, acc, lane, false);
  }
  __syncthreads();
  // ---- cross-entropy loss + accuracy, one thread per row
  if (tid < MB) {
    const int row = tid;
    const float* byr = by + (size_t)(b0 + row) * C_N;
    float mx = zs[row][0];
    int am = 0;
    for (int cc = 1; cc < C_N; ++cc) {
      const float v = zs[row][cc];
      if (v > mx) { mx = v; am = cc; }
    }
    float se = 0.f;
    for (int cc = 0; cc < C_N; ++cc) se += expf(zs[row][cc] - mx);
    const float lse = logf(se) + mx;
    float loss = 0.f, mxy = byr[0];
    int amy = 0;
    for (int cc = 0; cc < C_N; ++cc) {
      const float yv = byr[cc];
      loss += yv * (lse - zs[row][cc]);
      if (cc > 0 && yv > mxy) { mxy = yv; amy = cc; }
    }
    atomicAdd(&out[0], loss * (1.0f / B_N));
    atomicAdd(&out[1], (am == amy) ? (1.0f / B_N) : 0.0f);
  }
}

extern "C" void kernel_launch(void* const* d_in, const int* in_sizes, int n_in,
                              void* d_out, int out_size, void* d_ws, size_t ws_size,
                              hipStream_t stream) {
  (void)in_sizes; (void)n_in; (void)out_size; (void)ws_size;
  const float* bx = (const float*)d_in[0];
  const float* by = (const float*)d_in[1];
  const float* w1 = (const float*)d_in[2];
  const float* b1 = (const float*)d_in[3];
  const float* w2 = (const float*)d_in[4];
  const float* b2 = (const float*)d_in[5];
  const float* wr = (const float*)d_in[6];
  const float* cm = (const float*)d_in[7];
  const float* gv = (const float*)d_in[8];
  const float* bv = (const float*)d_in[9];
  const float* w3 = (const float*)d_in[10];
  const float* b3 = (const float*)d_in[11];
  const float* w4 = (const float*)d_in[12];
  const float* b4 = (const float*)d_in[13];
  float* out = (float*)d_out;
  float* ws = (float*)d_ws;

  float* a_state = ws;                                    // 2048*64*64 = 8,388,608 f32
  float* partial = ws + (size_t)B_N * H_N * H_N;          // 2*64*128   =    16,384 f32
  unsigned* bar  = (unsigned*)(partial + 2 * NBLK * 128); // 2 u32

  fw_init<<<1, 64, 0, stream>>>(bar, out);
  fw_main<<<NBLK, NTHR, 0, stream>>>(bx, by, w1, b1, w2, b2, wr, cm, gv, bv,
                                     w3, b3, w4, b4, a_state, partial, bar, out);
}